// GraphReadout_14104672600322
// MI455X (gfx1250) — compile-verified
//
#include <hip/hip_runtime.h>
#include <math.h>

typedef __attribute__((ext_vector_type(2))) float v2f;
typedef __attribute__((ext_vector_type(8))) float v8f;

#define D 256
#define H 4

__device__ __forceinline__ int lower_bound_i(const int* a, int n, int key) {
  int lo = 0, hi = n;
  while (lo < hi) { int mid = (lo + hi) >> 1; if (a[mid] < key) lo = mid + 1; else hi = mid; }
  return lo;
}

__device__ __forceinline__ float gelu_exact(float v) {
  return 0.5f * v * (1.0f + erff(v * 0.70710678118654752f));
}

// ---------------------------------------------------------------------------
// Kernel 1: one pass over x. membership is sorted -> each segment is a
// contiguous row range. One block per segment; thread t owns feature column t.
// Computes: segment sum/max of gelu(x), segment sum/max of x, and the
// attention branch with an ONLINE softmax (running max/sum + rescaled
// weighted accumulator) so x is read exactly once (512 MB total).
// ---------------------------------------------------------------------------
__global__ __launch_bounds__(256) void seg_stream_kernel(
    const float* __restrict__ x, const int* __restrict__ seg,
    const float* __restrict__ att_w, int N,
    float* __restrict__ center, float* __restrict__ neighbor,
    float* __restrict__ att, float* __restrict__ maxo, float* __restrict__ sumo)
{
  const int b    = blockIdx.x;
  const int t    = threadIdx.x;      // feature column 0..255
  const int wave = t >> 5;
  const int lane = t & 31;
  const int ht   = t >> 6;           // head of this column (64 cols per head)

  __shared__ float part[8][8];       // [wave][row] partial head-dot sums
  __shared__ float e_lds[8][4];      // [row][head] exp weights (shifted)
  __shared__ float scale_lds[4];     // per-head rescale for accumulators
  __shared__ float m_lds[4];         // running max per head
  __shared__ float s_lds[4];         // running sum per head

  if (t < 4) { m_lds[t] = -INFINITY; s_lds[t] = 0.0f; }
  __syncthreads();

  const int start = lower_bound_i(seg, N, b);
  const int end   = lower_bound_i(seg, N, b + 1);

  const float w_t = att_w[t];        // att_w flat (H, D/H) == column-indexed
  float sum_x = 0.f, sum_g = 0.f, acc = 0.f;
  float max_x = -INFINITY, max_g = -INFINITY;

  for (int i0 = start; i0 < end; i0 += 8) {
    const int R = min(8, end - i0);
    float v[8];
#pragma unroll
    for (int r = 0; r < 8; ++r) {
      float val = 0.f;
      if (r < R) {
        val = x[(size_t)(i0 + r) * D + t];
        sum_x += val;
        max_x = fmaxf(max_x, val);
        const float g = gelu_exact(val);
        sum_g += g;
        max_g = fmaxf(max_g, g);
      }
      v[r] = val;
    }
    if (i0 + 8 < end) __builtin_prefetch(&x[(size_t)(i0 + 8) * D + t], 0, 1);

    // per-wave (32-column) partial of the head dot product, per row
#pragma unroll
    for (int r = 0; r < 8; ++r) {
      float p = v[r] * w_t;
#pragma unroll
      for (int off = 16; off >= 1; off >>= 1) p += __shfl_xor(p, off, 32);
      if (lane == 0) part[wave][r] = p;
    }
    __syncthreads();

    // wave 0: finish alpha (head h = waves 2h,2h+1), online-softmax update
    if (t < 32) {
      const int r = t >> 2, h = t & 3;
      float a = part[2 * h][r] + part[2 * h + 1][r];
      a = (a > 0.f) ? a : 0.2f * a;                     // leaky_relu(0.2)
      const float al = (r < R) ? a : -INFINITY;
      float cm = al;
#pragma unroll
      for (int off = 4; off <= 16; off <<= 1) cm = fmaxf(cm, __shfl_xor(cm, off, 32));
      const float mo = m_lds[h];
      const float nm = fmaxf(mo, cm);
      const float sc = expf(mo - nm);                   // 0 on first chunk
      const float e  = (r < R) ? expf(al - nm) : 0.f;
      float cs = e;
#pragma unroll
      for (int off = 4; off <= 16; off <<= 1) cs += __shfl_xor(cs, off, 32);
      e_lds[r][h] = e;
      if (r == 0) {
        s_lds[h]     = s_lds[h] * sc + cs;
        m_lds[h]     = nm;
        scale_lds[h] = sc;
      }
    }
    __syncthreads();

    // all threads: rescale weighted accumulator, add this chunk
    const float sc = scale_lds[ht];
    float add = 0.f;
#pragma unroll
    for (int r = 0; r < 8; ++r) add += e_lds[r][ht] * v[r];
    acc = acc * sc + add;
    __syncthreads();
  }

  const float denom = s_lds[ht];
  const size_t o = (size_t)b * D + t;
  center[o]   = sum_g;               // segment_sum(gelu(x))
  neighbor[o] = max_g;               // segment_max(gelu(x))
  sumo[o]     = sum_x;
  maxo[o]     = max_x;
  att[o]      = (denom > 0.f) ? (acc / denom) : 0.f;
}

// ---------------------------------------------------------------------------
// Kernel 2: gi = center @ W_ih^T + b_ih, gh = neighbor @ W_hh^T + b_hh,
// fused GRU gating -> rnn_out (B, D). One 16-row m-tile per block; each wave
// owns two 16-col d-tiles and accumulates the 6 required GEMM tiles
// (i_r,i_z,i_n,h_r,h_z,h_n) with V_WMMA_F32_16X16X4_F32 (K-loop of 64).
// ---------------------------------------------------------------------------
__global__ __launch_bounds__(256) void gru_gemm_kernel(
    const float* __restrict__ center, const float* __restrict__ neighbor,
    const float* __restrict__ w_ih, const float* __restrict__ w_hh,
    const float* __restrict__ b_ih, const float* __restrict__ b_hh,
    float* __restrict__ rnn)
{
  const int lane = threadIdx.x & 31;
  const int wave = threadIdx.x >> 5;
  const int m0   = blockIdx.x * 16;
  const int aRow = m0 + (lane & 15);          // A: lanes 0-15 M, 16-31 same M
  const int kOff = (lane >> 4) << 1;          // upper half-lanes take K+2,K+3
  const int bL   = lane & 15;                 // B column within tile

  const float* cA = center   + (size_t)aRow * D;
  const float* nA = neighbor + (size_t)aRow * D;

  for (int dt = wave; dt < 16; dt += 8) {
    const int n0 = dt * 16;
    v8f a0 = {}, a1 = {}, a2 = {}, a3 = {}, a4 = {}, a5 = {};
    const float* bi0 = w_ih + (size_t)(n0 +       bL) * D;
    const float* bi1 = w_ih + (size_t)(n0 + 256 + bL) * D;
    const float* bi2 = w_ih + (size_t)(n0 + 512 + bL) * D;
    const float* bh0 = w_hh + (size_t)(n0 +       bL) * D;
    const float* bh1 = w_hh + (size_t)(n0 + 256 + bL) * D;
    const float* bh2 = w_hh + (size_t)(n0 + 512 + bL) * D;

    for (int k0 = 0; k0 < D; k0 += 4) {
      const int k = k0 + kOff;
      v2f ac, an, v0, v1, v2, u0, u1, u2;
      ac[0] = cA[k];  ac[1] = cA[k + 1];
      an[0] = nA[k];  an[1] = nA[k + 1];
      v0[0] = bi0[k]; v0[1] = bi0[k + 1];
      v1[0] = bi1[k]; v1[1] = bi1[k + 1];
      v2[0] = bi2[k]; v2[1] = bi2[k + 1];
      u0[0] = bh0[k]; u0[1] = bh0[k + 1];
      u1[0] = bh1[k]; u1[1] = bh1[k + 1];
      u2[0] = bh2[k]; u2[1] = bh2[k + 1];
      a0 = __builtin_amdgcn_wmma_f32_16x16x4_f32(false, ac, false, v0, (short)0, a0, false, false);
      a1 = __builtin_amdgcn_wmma_f32_16x16x4_f32(false, ac, false, v1, (short)0, a1, false, false);
      a2 = __builtin_amdgcn_wmma_f32_16x16x4_f32(false, ac, false, v2, (short)0, a2, false, false);
      a3 = __builtin_amdgcn_wmma_f32_16x16x4_f32(false, an, false, u0, (short)0, a3, false, false);
      a4 = __builtin_amdgcn_wmma_f32_16x16x4_f32(false, an, false, u1, (short)0, a4, false, false);
      a5 = __builtin_amdgcn_wmma_f32_16x16x4_f32(false, an, false, u2, (short)0, a5, false, false);
    }

    // fused GRU gating on the 16x16 tile (C/D layout: VGPR i -> M = i (+8))
    const int col = n0 + bL;
    const float bir = b_ih[col], biz = b_ih[col + 256], bin_ = b_ih[col + 512];
    const float bhr = b_hh[col], bhz = b_hh[col + 256], bhn  = b_hh[col + 512];
    const int rbase = m0 + ((lane >> 4) << 3);
#pragma unroll
    for (int i = 0; i < 8; ++i) {
      const int row = rbase + i;
      const float ir = a0[i] + bir, iz = a1[i] + biz, in_ = a2[i] + bin_;
      const float hr = a3[i] + bhr, hz = a4[i] + bhz, hn  = a5[i] + bhn;
      const float r  = 1.f / (1.f + expf(-(ir + hr)));
      const float z  = 1.f / (1.f + expf(-(iz + hz)));
      const float ng = tanhf(in_ + r * hn);
      const float nb = neighbor[(size_t)row * D + col];
      rnn[(size_t)row * D + col] = (1.f - z) * ng + z * nb;
    }
  }
}

// ---------------------------------------------------------------------------
// Kernel 3: out = [rnn|att|max|sum] @ merge_w^T + merge_b, without ever
// materializing the concatenation: the four K=256 blocks come straight from
// the workspace arrays. WMMA f32 16x16x4, K-loop of 256, fused bias.
// ---------------------------------------------------------------------------
__global__ __launch_bounds__(256) void merge_gemm_kernel(
    const float* __restrict__ rnn, const float* __restrict__ att,
    const float* __restrict__ maxo, const float* __restrict__ sumo,
    const float* __restrict__ merge_w, const float* __restrict__ merge_b,
    float* __restrict__ out)
{
  const int lane = threadIdx.x & 31;
  const int wave = threadIdx.x >> 5;
  const int m0   = blockIdx.x * 16;
  const int aRow = m0 + (lane & 15);
  const int kOff = (lane >> 4) << 1;
  const int bL   = lane & 15;
  const float* srcs[4] = { rnn, att, maxo, sumo };

  for (int nt = wave; nt < 16; nt += 8) {
    const int n0 = nt * 16;
    v8f acc = {};
    const float* wrow = merge_w + (size_t)(n0 + bL) * (4 * D);
#pragma unroll
    for (int s = 0; s < 4; ++s) {
      const float* A  = srcs[s] + (size_t)aRow * D;
      const float* Wp = wrow + s * D;
      for (int k0 = 0; k0 < D; k0 += 4) {
        const int k = k0 + kOff;
        v2f a, bv;
        a[0]  = A[k];  a[1]  = A[k + 1];
        bv[0] = Wp[k]; bv[1] = Wp[k + 1];
        acc = __builtin_amdgcn_wmma_f32_16x16x4_f32(false, a, false, bv, (short)0, acc, false, false);
      }
    }
    const int col  = n0 + bL;
    const float bb = merge_b[col];
    const int rbase = m0 + ((lane >> 4) << 3);
#pragma unroll
    for (int i = 0; i < 8; ++i)
      out[(size_t)(rbase + i) * D + col] = acc[i] + bb;
  }
}

// ---------------------------------------------------------------------------
extern "C" void kernel_launch(void* const* d_in, const int* in_sizes, int n_in,
                              void* d_out, int out_size, void* d_ws, size_t ws_size,
                              hipStream_t stream) {
  const float* x       = (const float*)d_in[0];
  const int*   seg     = (const int*)  d_in[1];
  const float* att_w   = (const float*)d_in[2];
  const float* w_ih    = (const float*)d_in[3];
  const float* w_hh    = (const float*)d_in[4];
  const float* b_ih    = (const float*)d_in[5];
  const float* b_hh    = (const float*)d_in[6];
  const float* merge_w = (const float*)d_in[7];
  const float* merge_b = (const float*)d_in[8];
  float* out = (float*)d_out;

  const int N = in_sizes[1];          // 500000 nodes
  const int B = out_size / D;         // 4096 segments
  const size_t BD = (size_t)B * D;

  float* wsf      = (float*)d_ws;     // 6 * 4 MB = 24 MB scratch
  float* center   = wsf;
  float* neighbor = wsf + 1 * BD;
  float* att      = wsf + 2 * BD;
  float* maxo     = wsf + 3 * BD;
  float* sumo     = wsf + 4 * BD;
  float* rnn      = wsf + 5 * BD;

  seg_stream_kernel<<<B, 256, 0, stream>>>(x, seg, att_w, N,
                                           center, neighbor, att, maxo, sumo);
  gru_gemm_kernel<<<B / 16, 256, 0, stream>>>(center, neighbor, w_ih, w_hh,
                                              b_ih, b_hh, rnn);
  merge_gemm_kernel<<<B / 16, 256, 0, stream>>>(rnn, att, maxo, sumo,
                                                merge_w, merge_b, out);
}